// BinsCombinerLayer_7164005450244
// MI455X (gfx1250) — compile-verified
//
#include <hip/hip_runtime.h>

// CDNA5 / gfx1250. wave32. Uses V_WMMA_F32_16X16X4_F32 for the final
// cross-lane row-sum reduction (B = all-ones -> D[m,n] = sum_k A[m,k]).

typedef float v2f __attribute__((ext_vector_type(2)));
typedef float v8f __attribute__((ext_vector_type(8)));

#define BATCH            32768
#define KTOT             2048     // 16 bins * 128 = floats per example row
#define ROWS_PER_WAVE    16
#define WAVES_PER_BLOCK  8
#define THREADS          (WAVES_PER_BLOCK * 32)
#define ROWS_PER_BLOCK   (ROWS_PER_WAVE * WAVES_PER_BLOCK)   // 128
#define NBLOCKS          (BATCH / ROWS_PER_BLOCK)            // 256

__global__ __launch_bounds__(THREADS)
void bins_combiner_kernel(const float* __restrict__ in,
                          const float* __restrict__ cent,
                          float* __restrict__ out)
{
    __shared__ float c_lds[KTOT];                    // 8 KB centroids
    __shared__ float red[WAVES_PER_BLOCK * 32 * 16]; // 16 KB partial sums

    const int tid  = threadIdx.x;
    const int lane = tid & 31;
    const int wave = tid >> 5;

    // ---- stage centroids into LDS: 2048 floats = 512 float4, 2 per thread
    {
        const float4* c4 = (const float4*)cent;
        float4*       s4 = (float4*)c_lds;
        s4[tid]       = c4[tid];
        s4[tid + 256] = c4[tid + 256];
    }
    __syncthreads();

    // ---- streaming phase: each wave owns 16 consecutive rows (8 KB each)
    const int rowBase = blockIdx.x * ROWS_PER_BLOCK + wave * ROWS_PER_WAVE;
    const float4* __restrict__ in4 = (const float4*)(in + (size_t)rowBase * KTOT);
    const float4* __restrict__ c4  = (const float4*)c_lds;

    float acc[16];
    #pragma unroll
    for (int r = 0; r < 16; ++r) acc[r] = 0.0f;

    // k is swept in 16 chunks of 128 floats; lane l covers k = t*128 + 4l .. +3.
    for (int t = 0; t < 16; ++t) {
        const float4 cv = c4[t * 32 + lane];              // shared by all 16 rows
        if (t < 15)
            __builtin_prefetch(&in4[(t + 1) * 32 + lane], 0, 1); // global_prefetch_b8
        #pragma unroll
        for (int r = 0; r < 16; ++r) {
            const float4 x = in4[r * 512 + t * 32 + lane]; // coalesced 512B/wave
            acc[r] = fmaf(x.x, cv.x, acc[r]);
            acc[r] = fmaf(x.y, cv.y, acc[r]);
            acc[r] = fmaf(x.z, cv.z, acc[r]);
            acc[r] = fmaf(x.w, cv.w, acc[r]);
        }
    }

    // ---- transpose partials through LDS: red[wave][lane][row]
    float* redw = red + wave * 512;
    {
        float4* dst = (float4*)(redw + lane * 16);
        dst[0] = make_float4(acc[0],  acc[1],  acc[2],  acc[3]);
        dst[1] = make_float4(acc[4],  acc[5],  acc[6],  acc[7]);
        dst[2] = make_float4(acc[8],  acc[9],  acc[10], acc[11]);
        dst[3] = make_float4(acc[12], acc[13], acc[14], acc[15]);
    }
    __syncthreads();

    // ---- WMMA reduction: S[m][j] = partial(row m, lane j), 16x32 matrix.
    // 8 tiles of 16x4; A layout (ISA 7.12.2): lane L: v0=K0, v1=K1;
    // lane L+16: v0=K2, v1=K3. B = all ones -> D[m,*] = row sums, chained in C.
    const int m    = lane & 15;
    const int half = (lane >> 4) << 1;   // 0 for lanes 0-15, 2 for lanes 16-31
    v8f dacc = {0.f, 0.f, 0.f, 0.f, 0.f, 0.f, 0.f, 0.f};
    const v2f ones = {1.0f, 1.0f};
    #pragma unroll
    for (int u = 0; u < 8; ++u) {
        v2f a;
        a[0] = redw[(4 * u + half    ) * 16 + m];  // S[m][4u + half]
        a[1] = redw[(4 * u + half + 1) * 16 + m];  // S[m][4u + half + 1]
        dacc = __builtin_amdgcn_wmma_f32_16x16x4_f32(
            /*neg_a=*/false, a, /*neg_b=*/false, ones,
            /*c_mod=*/(short)0, dacc, /*reuse_a=*/false, /*reuse_b=*/false);
    }

    // ---- D layout: reg p, lanes 0-15 -> row p, lanes 16-31 -> row p+8.
    // Lane 0 writes rows 0-7, lane 16 writes rows 8-15 (contiguous float4s).
    if (m == 0) {
        const float s = 1.0f / 16.0f;   // mean over bins
        float4* o4 = (float4*)(out + rowBase + ((lane >> 4) << 3));
        o4[0] = make_float4(dacc[0] * s, dacc[1] * s, dacc[2] * s, dacc[3] * s);
        o4[1] = make_float4(dacc[4] * s, dacc[5] * s, dacc[6] * s, dacc[7] * s);
    }
}

extern "C" void kernel_launch(void* const* d_in, const int* in_sizes, int n_in,
                              void* d_out, int out_size, void* d_ws, size_t ws_size,
                              hipStream_t stream) {
    const float* in   = (const float*)d_in[0];   // [32768, 16, 128] fp32
    const float* cent = (const float*)d_in[1];   // [16, 128] fp32
    float*       out  = (float*)d_out;           // [32768] fp32
    (void)in_sizes; (void)n_in; (void)out_size; (void)d_ws; (void)ws_size;
    bins_combiner_kernel<<<dim3(NBLOCKS), dim3(THREADS), 0, stream>>>(in, cent, out);
}